// TopKPoolWeighted_88656714924079
// MI455X (gfx1250) — compile-verified
//
#include <hip/hip_runtime.h>
#include <math.h>

#define B_ 64
#define N_ 1024
#define C_ 128
#define K_ 512   // ceil(0.5 * 1024)

typedef __attribute__((ext_vector_type(2))) float v2f;
typedef __attribute__((ext_vector_type(8))) float v8f;

// ---------------------------------------------------------------------------
// Kernel 1: score[b,n] = tanh( (x[b,n,:] . w) / ||w|| ) via V_WMMA_F32_16X16X4_F32
// One wave (32 threads) computes scores for a tile of 16 nodes, accumulating
// 32 WMMA steps over the C=128 channel dimension in exact f32.
// ---------------------------------------------------------------------------
__global__ __launch_bounds__(32)
void score_wmma_kernel(const float* __restrict__ x,
                       const float* __restrict__ w,
                       float* __restrict__ scores) {
  const int n0   = blockIdx.x * 16;
  const int b    = blockIdx.y;
  const int l    = threadIdx.x;     // 0..31, all lanes active (EXEC all-ones)
  const int half = l >> 4;          // 0 or 1
  const int m    = l & 15;          // node within tile / column id

  // ||w||^2, uniform across lanes (w is tiny, stays in cache)
  float s2 = 0.0f;
  for (int c = 0; c < C_; ++c) { float wv = w[c]; s2 += wv * wv; }
  const float nrm = sqrtf(s2);

  const float* xr = x + ((size_t)b * N_) * (size_t)C_;

  v8f acc = {};
  for (int kb = 0; kb < C_; kb += 4) {
    const int kA = kb + 2 * half;
    const float* xp = xr + (size_t)(n0 + m) * C_ + kA;
    v2f a;  a.x  = xp[0];  a.y  = xp[1];
    v2f bv; bv.x = w[kA];  bv.y = w[kA + 1];
    acc = __builtin_amdgcn_wmma_f32_16x16x4_f32(
        false, a, false, bv, (short)0, acc, false, false);
  }

  if (m == 0) {  // lanes 0 and 16 hold column N=0 for rows M = r + 8*half
    for (int r = 0; r < 8; ++r) {
      const int node = n0 + r + 8 * half;
      scores[(size_t)b * N_ + node] = tanhf(acc[r] / nrm);
    }
  }
}

// ---------------------------------------------------------------------------
// Kernel 2: per-batch top-k (k=512) of 1024 scores via bitonic sort in LDS of
// composite keys: high 32 bits = ~orderable(score) (descending score),
// low 32 bits = index (ascending tie-break) — matches jax.lax.top_k stability.
// ---------------------------------------------------------------------------
__global__ __launch_bounds__(512)
void topk_sort_kernel(const float* __restrict__ scores,
                      int* __restrict__ perm,
                      float* __restrict__ sel) {
  __shared__ unsigned long long keys[N_];
  const int b = blockIdx.x;
  const int t = threadIdx.x;   // 0..511

  for (int e = t; e < N_; e += 512) {
    float s = scores[(size_t)b * N_ + e];
    unsigned u = __float_as_uint(s);
    u = (u & 0x80000000u) ? ~u : (u | 0x80000000u);  // orderable ascending
    keys[e] = ((unsigned long long)(~u) << 32) | (unsigned)e;
  }

  for (int k2 = 2; k2 <= N_; k2 <<= 1) {
    for (int j = k2 >> 1; j > 0; j >>= 1) {
      __syncthreads();
      const int i  = 2 * t - (t & (j - 1));
      const int p  = i + j;
      const bool up = ((i & k2) == 0);
      unsigned long long a = keys[i], c = keys[p];
      if ((a > c) == up) { keys[i] = c; keys[p] = a; }
    }
  }
  __syncthreads();

  if (t < K_) {
    const unsigned idx = (unsigned)keys[t];
    perm[(size_t)b * K_ + t] = (int)idx;
    sel [(size_t)b * K_ + t] = scores[(size_t)b * N_ + idx];
  }
}

// ---------------------------------------------------------------------------
// Kernel 3: x_pool[b,i,c] = x[b, perm[i], c] * sel_score[b,i]   (coalesced)
// ---------------------------------------------------------------------------
__global__ __launch_bounds__(128)
void xpool_kernel(const float* __restrict__ x,
                  const int* __restrict__ perm,
                  const float* __restrict__ sel,
                  float* __restrict__ out) {
  const int i = blockIdx.x, b = blockIdx.y, c = threadIdx.x;
  const int   p = perm[(size_t)b * K_ + i];
  const float s = sel [(size_t)b * K_ + i];
  out[((size_t)b * K_ + i) * C_ + c] =
      x[((size_t)b * N_ + p) * C_ + c] * s;
}

// ---------------------------------------------------------------------------
// Kernel 4: adj_pooled[b,i,j] = ((adj[b,pi,pj]!=0) || (adj[b,pj,pi]!=0)) + (i==j)
// Row p_i (4 KB) is staged into LDS with GLOBAL_LOAD_ASYNC_TO_LDS_B64
// (ASYNCcnt path, contiguous full-line stream), then gathered from LDS.
// The transpose term rides the 192 MB L2 (4 MB adj slice per batch).
// ---------------------------------------------------------------------------
__global__ __launch_bounds__(512)
void adjpool_kernel(const float* __restrict__ adj,
                    const int* __restrict__ perm,
                    float* __restrict__ out) {
  __shared__ int   ps[K_];
  __shared__ float rowLds[N_];

  const int i = blockIdx.x, b = blockIdx.y, j = threadIdx.x;
  ps[j] = perm[(size_t)b * K_ + j];
  __syncthreads();

  const int pi = ps[i], pj = ps[j];
  const float* ab    = adj + (size_t)b * N_ * N_;
  const float* rowPi = ab + (size_t)pi * N_;
  const float* rowT  = ab + (size_t)pj * N_;

  // Async-stage row p_i into LDS: each thread moves 8 bytes (2 floats).
  {
    unsigned           ldsAddr = (unsigned)(size_t)(&rowLds[2 * j]); // LDS byte offset (low 32 bits of flat shared ptr)
    unsigned long long gAddr   = (unsigned long long)(size_t)(rowPi + 2 * j);
    asm volatile("global_load_async_to_lds_b64 %0, %1, off"
                 :: "v"(ldsAddr), "v"(gAddr) : "memory");
    asm volatile("s_wait_asynccnt 0x0" ::: "memory");
  }
  __builtin_prefetch(rowT + pi, 0, 3);               // global_prefetch_b8
  __syncthreads();                                    // row visible to all waves

  const float v1 = rowLds[pj];     // adj[b, pi, pj] via LDS gather
  const float v2 = rowT[pi];       // adj[b, pj, pi] via L2
  float o = ((v1 != 0.0f) || (v2 != 0.0f)) ? 1.0f : 0.0f;
  if (j == i) o += 1.0f;
  out[((size_t)b * K_ + i) * K_ + j] = o;
}

// ---------------------------------------------------------------------------
extern "C" void kernel_launch(void* const* d_in, const int* in_sizes, int n_in,
                              void* d_out, int out_size, void* d_ws, size_t ws_size,
                              hipStream_t stream) {
  const float* x   = (const float*)d_in[0];  // [B,N,C]
  const float* adj = (const float*)d_in[1];  // [B,N,N]
  const float* w   = (const float*)d_in[2];  // [C]

  float* xpool = (float*)d_out;                              // B*K*C floats
  float* adjp  = (float*)d_out + (size_t)B_ * K_ * C_;       // B*K*K floats

  char*  ws     = (char*)d_ws;
  float* scores = (float*)ws;                                           // B*N f32
  int*   perm   = (int*)  (ws + (size_t)B_ * N_ * sizeof(float));       // B*K i32
  float* sel    = (float*)(ws + (size_t)B_ * N_ * sizeof(float)
                              + (size_t)B_ * K_ * sizeof(int));         // B*K f32

  score_wmma_kernel<<<dim3(N_ / 16, B_), 32,  0, stream>>>(x, w, scores);
  topk_sort_kernel <<<dim3(B_),         512, 0, stream>>>(scores, perm, sel);
  xpool_kernel     <<<dim3(K_, B_),     C_,  0, stream>>>(x, perm, sel, xpool);
  adjpool_kernel   <<<dim3(K_, B_),     K_,  0, stream>>>(adj, perm, adjp);
}